// BboxLayer_36893769073007
// MI455X (gfx1250) — compile-verified
//
#include <hip/hip_runtime.h>
#include <hip/hip_bf16.h>
#include <stdint.h>

#define HH_IMG 1536
#define WW_IMG 2048
#define NPIX (HH_IMG * WW_IMG)        // 3145728 = 12288 * 256
#define SENT NPIX                     // background sentinel label
#define NB_WORDS ((NPIX + 31) / 32)   // text-flag bitmask words

// ---------------- union-find helpers (agent-coherent) ----------------

__device__ __forceinline__ int ld_agent(const int* p) {
    return __hip_atomic_load(p, __ATOMIC_RELAXED, __HIP_MEMORY_SCOPE_AGENT);
}

__device__ __forceinline__ int find_root(int* L, int x) {
    int p = ld_agent(L + x);
    while (p != x) { x = p; p = ld_agent(L + x); }
    return x;
}

// Lock-free union by atomicMin: larger root is always hung under smaller
// value, so the final root of a component is its minimum flat pixel index
// (matches the reference's min-label propagation exactly).
__device__ void merge_sets(int* L, int a, int b) {
    a = find_root(L, a);
    b = find_root(L, b);
    while (a != b) {
        if (a < b) { int t = a; a = b; b = t; }   // ensure a > b
        int old = atomicMin(&L[a], b);
        if (old == a) break;                       // a was root, now child of b
        a = old;                                   // follow existing parent (old < a)
    }
}

// ---------------- kernel 1: threshold mask + stats init ----------------
// mask bit0 = comb = (x0>0.4 || x1>0.4), bit1 = text = (x0>0.4)

__global__ void __launch_bounds__(256)
k_mask_init(const float2* __restrict__ x, unsigned char* __restrict__ mask,
            int* __restrict__ ymin, int* __restrict__ ymax,
            int* __restrict__ xmin, int* __restrict__ xmax,
            unsigned int* __restrict__ tbits) {
    int i = blockIdx.x * 256 + threadIdx.x;
    if (i >= NPIX) return;
    float2 v = x[i];
    int text = (v.x > 0.4f) ? 1 : 0;
    int comb = ((v.x > 0.4f) || (v.y > 0.4f)) ? 1 : 0;
    mask[i] = (unsigned char)(comb | (text << 1));
    ymin[i] = 0x7fffffff;  xmin[i] = 0x7fffffff;
    ymax[i] = -1;          xmax[i] = -1;
    if (i < NB_WORDS) tbits[i] = 0u;
}

// ---------------- kernel 2: 3x3 dilation + run-based label init ----------------
// Tile 32x8 outputs per block (one row per wave32); 34x10 byte halo staged
// into LDS with gfx1250 async-to-LDS loads (ASYNCcnt path).

#define TW 32
#define THT 8
#define HLW (TW + 2)    // 34
#define HLH (THT + 2)   // 10

__global__ void __launch_bounds__(256)
k_dilate_label(const unsigned char* __restrict__ mask, int* __restrict__ label) {
    __shared__ unsigned char tile[HLH * HLW];   // 340 B
    const int tid = threadIdx.x;
    const int bx = blockIdx.x * TW;
    const int by = blockIdx.y * THT;

    // zero halo (covers out-of-image border)
    for (int k = tid; k < HLH * HLW; k += 256) tile[k] = 0;
    __syncthreads();

    // async-stage halo bytes: global -> LDS, per-lane addresses
    for (int k = tid; k < HLH * HLW; k += 256) {
        int hy = k / HLW, hx = k - hy * HLW;
        int gy = by + hy - 1, gx = bx + hx - 1;
        if ((unsigned)gx < (unsigned)WW_IMG && (unsigned)gy < (unsigned)HH_IMG) {
            unsigned lds_addr = (unsigned)(size_t)(&tile[k]);
            unsigned long long gaddr =
                (unsigned long long)(size_t)(mask + (size_t)gy * WW_IMG + gx);
            asm volatile("global_load_async_to_lds_b8 %0, %1, off"
                         :: "v"(lds_addr), "v"(gaddr) : "memory");
        }
    }
    asm volatile("s_wait_asynccnt 0x0" ::: "memory");
    __syncthreads();

    const int lane = tid & 31;   // x within tile (one wave == one tile row)
    const int ty   = tid >> 5;   // y within tile
    const int ox = bx + lane;
    const int oy = by + ty;

    int fg = 0;
#pragma unroll
    for (int dy = 0; dy < 3; ++dy) {
        int ro = (ty + dy) * HLW + lane;
        fg |= tile[ro] | tile[ro + 1] | tile[ro + 2];
    }
    fg &= 1;   // bit0 = comb; 3x3 OR == binary dilation

    // per-row run representative: leftmost fg pixel of this lane's run
    unsigned m = __builtin_amdgcn_ballot_w32(fg != 0);
    unsigned below_bg = (~m) & ((1u << lane) - 1u);
    int start = below_bg ? (32 - __clz(below_bg)) : 0;

    int i = oy * WW_IMG + ox;
    label[i] = fg ? (oy * WW_IMG + bx + start) : SENT;
}

// ---------------- kernel 3: boundary merges (4-connectivity) ----------------
// Within-tile runs already share a label. Only merge:
//   left  : at 32-wide tile boundaries
//   up    : at the leftmost column of each (run, up-run) overlap

__global__ void __launch_bounds__(256)
k_merge(int* __restrict__ label) {
    int i = blockIdx.x * 256 + threadIdx.x;
    if (i >= NPIX) return;
    if (label[i] >= SENT) return;                 // background (stable test)
    int y = i / WW_IMG, x = i - y * WW_IMG;
    bool tileB  = (x & 31) == 0;
    bool leftFg = (x > 0) && (label[i - 1] < SENT);
    if (tileB && x > 0 && leftFg) merge_sets(label, i, i - 1);
    if (y > 0 && label[i - WW_IMG] < SENT) {
        bool iStart  = tileB || !leftFg;                          // my run starts here
        bool upStart = tileB || (label[i - WW_IMG - 1] >= SENT);  // up-run starts here
        if (iStart || upStart) merge_sets(label, i, i - WW_IMG);
    }
}

// ---------------- kernel 4: flatten (full path compression) ----------------

__global__ void __launch_bounds__(256)
k_flatten(int* __restrict__ label) {
    int i = blockIdx.x * 256 + threadIdx.x;
    if (i >= NPIX) return;
    int l = label[i];
    if (l >= SENT) return;
    label[i] = find_root(label, l);
}

// ---------------- kernel 5: per-component reductions ----------------
// Block-aggregate for the dominant root (handles the ~single giant
// component case with ~5 global atomics per 256 pixels).

__global__ void __launch_bounds__(256)
k_reduce(const int* __restrict__ label, const unsigned char* __restrict__ mask,
         int* __restrict__ ymin, int* __restrict__ ymax,
         int* __restrict__ xmin, int* __restrict__ xmax,
         unsigned int* __restrict__ tbits) {
    __shared__ int sdom;
    __shared__ int sred[4];   // ymin, ymax, xmin, xmax for dominant root
    __shared__ int stfl;
    int tid = threadIdx.x;
    int i = blockIdx.x * 256 + tid;

    int r = SENT;
    if (i < NPIX) r = label[i];
    bool fg = r < SENT;

    if (tid == 0) sdom = 0x7fffffff;
    __syncthreads();
    if (fg) atomicMin(&sdom, r);
    __syncthreads();
    int dom = sdom;
    if (tid == 0) { sred[0] = 0x7fffffff; sred[1] = -1;
                    sred[2] = 0x7fffffff; sred[3] = -1; stfl = 0; }
    __syncthreads();

    if (fg) {
        int y = i / WW_IMG, x = i - y * WW_IMG;
        int t = (mask[i] >> 1) & 1;
        if (r == dom) {
            atomicMin(&sred[0], y); atomicMax(&sred[1], y);
            atomicMin(&sred[2], x); atomicMax(&sred[3], x);
            if (t) atomicOr(&stfl, 1);
        } else {
            atomicMin(&ymin[r], y); atomicMax(&ymax[r], y);
            atomicMin(&xmin[r], x); atomicMax(&xmax[r], x);
            if (t) atomicOr(&tbits[r >> 5], 1u << (r & 31));
        }
    }
    __syncthreads();
    if (tid == 0 && dom != 0x7fffffff) {
        atomicMin(&ymin[dom], sred[0]); atomicMax(&ymax[dom], sred[1]);
        atomicMin(&xmin[dom], sred[2]); atomicMax(&xmax[dom], sred[3]);
        if (stfl) atomicOr(&tbits[dom >> 5], 1u << (dom & 31));
    }
}

// ---------------- kernel 6: emit bboxes + valid ----------------
// Position i is a nonempty-component root iff label[i] == i
// (background holds SENT != i) — exactly reference's count>0 at index i.

__global__ void __launch_bounds__(256)
k_output(const int* __restrict__ label,
         const int* __restrict__ ymin, const int* __restrict__ ymax,
         const int* __restrict__ xmin, const int* __restrict__ xmax,
         const unsigned int* __restrict__ tbits, float* __restrict__ out) {
    int i = blockIdx.x * 256 + threadIdx.x;
    if (i >= NPIX) return;
    float4 bb = make_float4(0.f, 0.f, 0.f, 0.f);
    float vv = 0.f;
    if (label[i] == i) {
        int y0 = ymin[i], y1 = ymax[i], x0 = xmin[i], x1 = xmax[i];
        int h = y1 - y0, w = x1 - x0;
        int t = (tbits[i >> 5] >> (i & 31)) & 1;
        if (h > 4 && w > 4 && t) {
            bb = make_float4((float)y0, (float)x0, (float)h, (float)w);
            vv = 1.f;
        }
    }
    ((float4*)out)[i] = bb;                       // bboxes[N,4]
    out[(size_t)4 * NPIX + i] = vv;               // valid[N]
}

// ---------------- launch ----------------

extern "C" void kernel_launch(void* const* d_in, const int* in_sizes, int n_in,
                              void* d_out, int out_size, void* d_ws, size_t ws_size,
                              hipStream_t stream) {
    const float2* x = (const float2*)d_in[0];
    char* ws = (char*)d_ws;
    const size_t A = (size_t)NPIX * 4;
    int*          label = (int*)(ws + 0 * A);
    int*          ymin  = (int*)(ws + 1 * A);
    int*          ymax  = (int*)(ws + 2 * A);
    int*          xmin  = (int*)(ws + 3 * A);
    int*          xmax  = (int*)(ws + 4 * A);
    unsigned int* tbits = (unsigned int*)(ws + 5 * A);
    unsigned char* mask = (unsigned char*)(ws + 5 * A + (size_t)NB_WORDS * 4);
    float* out = (float*)d_out;

    dim3 blk(256);
    dim3 g1(NPIX / 256);                          // 12288
    dim3 g2(WW_IMG / TW, HH_IMG / THT);           // 64 x 192

    k_mask_init   <<<g1, blk, 0, stream>>>(x, mask, ymin, ymax, xmin, xmax, tbits);
    k_dilate_label<<<g2, blk, 0, stream>>>(mask, label);
    k_merge       <<<g1, blk, 0, stream>>>(label);
    k_flatten     <<<g1, blk, 0, stream>>>(label);
    k_reduce      <<<g1, blk, 0, stream>>>(label, mask, ymin, ymax, xmin, xmax, tbits);
    k_output      <<<g1, blk, 0, stream>>>(label, ymin, ymax, xmin, xmax, tbits, out);
}